// LearnedImportanceSampler_24086176596699
// MI455X (gfx1250) — compile-verified
//
#include <hip/hip_runtime.h>
#include <hip/hip_bf16.h>
#include <math.h>

// Problem dims (match reference)
#define E_DIM  512
#define H_DIM  256
#define B_DIM  2
#define SQ_DIM 512
#define SK_DIM 512

// Padded LDS row stride (floats). 266*4 bytes is 8B-aligned; 266 % 64 == 10
// gives 16 distinct banks for the 16 concurrently-read A-fragment rows.
#define RS 266

typedef __attribute__((ext_vector_type(2))) float v2f;
typedef __attribute__((ext_vector_type(8))) float v8f;

// ---------------------------------------------------------------------------
// Prologue 1: feat = relu(x @ W + b)   x:[B*S,E]  W:[E,H]  out:[B*S,H]
// ---------------------------------------------------------------------------
__global__ void encode_kernel(const float* __restrict__ x, const float* __restrict__ W,
                              const float* __restrict__ bias, float* __restrict__ out) {
  int idx = blockIdx.x * blockDim.x + threadIdx.x;   // over B*S*H
  int h = idx & (H_DIM - 1);
  int s = idx >> 8;                                  // H_DIM == 256
  const float* xr = x + (long)s * E_DIM;
  float acc = bias[h];
#pragma unroll 4
  for (int e = 0; e < E_DIM; ++e) acc = fmaf(xr[e], W[e * H_DIM + h], acc);
  out[idx] = fmaxf(acc, 0.0f);
}

// ---------------------------------------------------------------------------
// Prologue 2: out = feat @ W[roff:roff+H, :] (+ bias)   feat:[B*S,H]  W:[2H,H]
// ---------------------------------------------------------------------------
__global__ void project_kernel(const float* __restrict__ feat, const float* __restrict__ W,
                               int roff, const float* __restrict__ bias,
                               float* __restrict__ out) {
  int idx = blockIdx.x * blockDim.x + threadIdx.x;   // over B*S*H
  int d = idx & (H_DIM - 1);
  int s = idx >> 8;
  const float* fr = feat + (long)s * H_DIM;
  float acc = bias ? bias[d] : 0.0f;
#pragma unroll 4
  for (int j = 0; j < H_DIM; ++j) acc = fmaf(fr[j], W[(roff + j) * H_DIM + d], acc);
  out[idx] = acc;
}

// ---------------------------------------------------------------------------
// One-shot weight packer: Wp[((nt*64 + kc)*32 + lane)*2 + v] =
//     W[(4*kc + (lane>>4)*2 + v)*H + nt*16 + (lane&15)]
// so each lane's WMMA B-fragment is one aligned 8-byte load, and a wave's
// fragment load touches one contiguous 256 B line.
// ---------------------------------------------------------------------------
__global__ void pack_w_kernel(const float* __restrict__ W, float* __restrict__ Wp) {
  int t = blockIdx.x * blockDim.x + threadIdx.x;   // [0, 16*64*32)
  int lane = t & 31;
  int kc   = (t >> 5) & 63;
  int nt   = t >> 11;
  int krow = (kc << 2) + ((lane >> 4) << 1);
  int ncol = (nt << 4) + (lane & 15);
  Wp[t * 2 + 0] = W[krow * H_DIM + ncol];
  Wp[t * 2 + 1] = W[(krow + 1) * H_DIM + ncol];
}

// ---------------------------------------------------------------------------
// One GEMM layer over a wave-private 32 x 256 LDS tile:
//   hout = relu(hin @ W + bias), all matrix math in v_wmma_f32_16x16x4_f32.
// Register blocking: 2 M-tiles x 8 N-tiles of accumulators (128 VGPRs);
// each packed B-fragment (global_load_b64) is reused by both M-tiles.
// ---------------------------------------------------------------------------
__device__ __forceinline__ void gemm_layer(const float* __restrict__ hin,
                                           float* __restrict__ hout,
                                           const float* __restrict__ Wp,
                                           const float* __restrict__ bias,
                                           int lane) {
  const int mrow  = lane & 15;          // A-fragment row within a 16-row tile
  const int khalf = (lane >> 4) << 1;   // K offset 0 or 2 within each K=4 chunk
  const int mbase = (lane >> 4) << 3;   // D-fragment M base (0 or 8)

  const float* Ab0 = hin + mrow * RS + khalf;
  const float* Ab1 = hin + (16 + mrow) * RS + khalf;

  for (int ntg = 0; ntg < 2; ++ntg) {
    v8f acc[2][8];
#pragma unroll
    for (int mt = 0; mt < 2; ++mt)
#pragma unroll
      for (int n = 0; n < 8; ++n) acc[mt][n] = (v8f){};

#pragma unroll 2
    for (int kc = 0; kc < 64; ++kc) {
      v2f a0, a1;
      a0.x = Ab0[4 * kc]; a0.y = Ab0[4 * kc + 1];
      a1.x = Ab1[4 * kc]; a1.y = Ab1[4 * kc + 1];
#pragma unroll
      for (int n = 0; n < 8; ++n) {
        const int nt = (ntg << 3) + n;
        v2f bb = *(const v2f*)(Wp + (size_t)((((nt << 6) + kc) << 5) + lane) * 2);
        acc[0][n] = __builtin_amdgcn_wmma_f32_16x16x4_f32(false, a0, false, bb,
                                                          (short)0, acc[0][n], false, false);
        acc[1][n] = __builtin_amdgcn_wmma_f32_16x16x4_f32(false, a1, false, bb,
                                                          (short)0, acc[1][n], false, false);
      }
    }

#pragma unroll
    for (int n = 0; n < 8; ++n) {
      const int nt = (ntg << 3) + n;
      const int ncol = (nt << 4) + mrow;
      const float bv = bias[ncol];
#pragma unroll
      for (int mt = 0; mt < 2; ++mt)
#pragma unroll
        for (int v = 0; v < 8; ++v)
          hout[((mt << 4) + mbase + v) * RS + ncol] = fmaxf(acc[mt][n][v] + bv, 0.0f);
    }
  }
}

// ---------------------------------------------------------------------------
// Main fused kernel. Block = 128 threads = 4 waves; each wave owns 32 pairs
// (one q, 32 consecutive k). The [32,256] activation tiles live only in LDS.
// ---------------------------------------------------------------------------
__global__ __launch_bounds__(128) void fused_pair_kernel(
    const float* __restrict__ qp,   // [B*SQ,H] = qf@W1[:H] + b1
    const float* __restrict__ kp,   // [B*SK,H] = kf@W1[H:]
    const float* __restrict__ vq,   // [B*SQ,H] = qf@Wv1[:H] + bv1
    const float* __restrict__ vk,   // [B*SK,H] = kf@Wv1[H:]
    const float* __restrict__ W2p, const float* __restrict__ b2,
    const float* __restrict__ W3p, const float* __restrict__ b3,
    const float* __restrict__ Wf,  const float* __restrict__ bf,
    const float* __restrict__ Wv2, const float* __restrict__ bv2,
    float* __restrict__ out_logits, float* __restrict__ out_var) {
  extern __shared__ float smem[];
  const int lane = threadIdx.x & 31;
  const int wave = threadIdx.x >> 5;
  const int bid  = blockIdx.x;
  const int kblk = bid & 3;                  // SK / (4 waves * 32) = 4 k-blocks
  const int q    = (bid >> 2) & (SQ_DIM - 1);
  const int batch = bid >> 11;               // / (4 * 512)
  const int k0 = (kblk << 7) + (wave << 5);  // this wave's first k (32 of them)

  float* h0 = smem + wave * (2 * 32 * RS);
  float* h1 = h0 + 32 * RS;

  const float* qrow  = qp + (long)(batch * SQ_DIM + q) * H_DIM;
  const float* krows = kp + (long)(batch * SK_DIM + k0) * H_DIM;

  // h0 = relu(qp[q] + kp[k0+m])   (b1 already folded into qp)
  for (int t = lane; t < 32 * H_DIM; t += 32) {
    int m = t >> 8;
    int j = t & (H_DIM - 1);
    h0[m * RS + j] = fmaxf(qrow[j] + krows[m * H_DIM + j], 0.0f);
  }
  __syncthreads();

  gemm_layer(h0, h1, W2p, b2, lane);   // h1 = relu(h0 @ W2 + b2)
  __syncthreads();
  gemm_layer(h1, h0, W3p, b3, lane);   // h3 = relu(h1 @ W3 + b3) -> h0 slice
  __syncthreads();

  const int mrow = lane & 15;
  const int joff = (lane >> 4) << 7;   // each half-wave covers 128 of H
  const long obase = (long)(batch * SQ_DIM + q) * SK_DIM + k0;

  // ---- logits epilogue: dot(h3[m,:], Wf) + bf ----
#pragma unroll
  for (int mt = 0; mt < 2; ++mt) {
    const float* row = h0 + ((mt << 4) + mrow) * RS + joff;
    const float* wf  = Wf + joff;
    float part = 0.0f;
#pragma unroll 4
    for (int j = 0; j < 128; ++j) part = fmaf(row[j], wf[j], part);
    part += __shfl_xor(part, 16, 32);
    if (lane < 16) out_logits[obase + (mt << 4) + mrow] = part + bf[0];
  }

  // ---- variance epilogue: softplus(dot(relu(vq+vk), Wv2) + bv2) ----
#pragma unroll
  for (int mt = 0; mt < 2; ++mt) {
    const float* vqrow = vq + (long)(batch * SQ_DIM + q) * H_DIM + joff;
    const float* vkrow = vk + (long)(batch * SK_DIM + k0 + (mt << 4) + mrow) * H_DIM + joff;
    const float* wv = Wv2 + joff;
    float part = 0.0f;
#pragma unroll 4
    for (int j = 0; j < 128; ++j) {
      float x = fmaxf(vqrow[j] + vkrow[j], 0.0f);
      part = fmaf(x, wv[j], part);
    }
    part += __shfl_xor(part, 16, 32);
    if (lane < 16) {
      float x = part + bv2[0];
      out_var[obase + (mt << 4) + mrow] = (x > 20.0f) ? x : log1pf(expf(x));
    }
  }
}

// ---------------------------------------------------------------------------
extern "C" void kernel_launch(void* const* d_in, const int* in_sizes, int n_in,
                              void* d_out, int out_size, void* d_ws, size_t ws_size,
                              hipStream_t stream) {
  (void)in_sizes; (void)n_in; (void)out_size; (void)ws_size;
  const float* query = (const float*)d_in[0];
  const float* key   = (const float*)d_in[1];
  const float* Wqe   = (const float*)d_in[2];
  const float* bqe   = (const float*)d_in[3];
  const float* Wke   = (const float*)d_in[4];
  const float* bke   = (const float*)d_in[5];
  const float* W1    = (const float*)d_in[6];
  const float* b1    = (const float*)d_in[7];
  const float* W2    = (const float*)d_in[8];
  const float* b2    = (const float*)d_in[9];
  const float* W3    = (const float*)d_in[10];
  const float* b3    = (const float*)d_in[11];
  const float* Wf    = (const float*)d_in[12];
  const float* bf    = (const float*)d_in[13];
  const float* Wv1   = (const float*)d_in[14];
  const float* bv1   = (const float*)d_in[15];
  const float* Wv2   = (const float*)d_in[16];
  const float* bv2   = (const float*)d_in[17];

  const size_t NSH = (size_t)B_DIM * SQ_DIM * H_DIM;   // 262144 floats per tensor
  float* qf  = (float*)d_ws;
  float* kf  = qf + NSH;
  float* qp  = kf + NSH;
  float* kp  = qp + NSH;
  float* vq  = kp + NSH;
  float* vk  = vq + NSH;
  float* W2p = vk + NSH;                 // 65536 floats
  float* W3p = W2p + (size_t)H_DIM * H_DIM;   // total workspace: 6.5 MB

  dim3 blk(256);
  dim3 grd((unsigned)(NSH / 256));   // 1024 blocks

  encode_kernel<<<grd, blk, 0, stream>>>(query, Wqe, bqe, qf);
  encode_kernel<<<grd, blk, 0, stream>>>(key,   Wke, bke, kf);
  project_kernel<<<grd, blk, 0, stream>>>(qf, W1,  0,     b1,      qp);
  project_kernel<<<grd, blk, 0, stream>>>(kf, W1,  H_DIM, nullptr, kp);
  project_kernel<<<grd, blk, 0, stream>>>(qf, Wv1, 0,     bv1,     vq);
  project_kernel<<<grd, blk, 0, stream>>>(kf, Wv1, H_DIM, nullptr, vk);
  pack_w_kernel<<<dim3(128), blk, 0, stream>>>(W2, W2p);
  pack_w_kernel<<<dim3(128), blk, 0, stream>>>(W3, W3p);

  // 4 waves * 2 tiles * 32 rows * 266 floats * 4 B = 272,384 B (< 320 KB WGP LDS)
  const size_t shmem = (size_t)(4 * 2 * 32 * RS) * sizeof(float);
  hipFuncSetAttribute(reinterpret_cast<const void*>(&fused_pair_kernel),
                      hipFuncAttributeMaxDynamicSharedMemorySize, (int)shmem);

  float* out_logits = (float*)d_out;
  float* out_var    = out_logits + (size_t)B_DIM * SQ_DIM * SK_DIM;
  fused_pair_kernel<<<dim3(B_DIM * SQ_DIM * 4), dim3(128), shmem, stream>>>(
      qp, kp, vq, vk, W2p, b2, W3p, b3, Wf, bf, Wv2, bv2, out_logits, out_var);
}